// MLA_28329604284808
// MI455X (gfx1250) — compile-verified
//
#include <hip/hip_runtime.h>

// ---------------------------------------------------------------------------
// MLA (DeepSeek-style) decode step on MI455X (gfx1250).
// Roofline: ~48 GFLOP vs ~0.85 GB of fp32 weights/caches -> HBM-bound at
// 23.3 TB/s (~37us). Keep the fp32 data path; V_WMMA_F32_16X16X4_F32 for all
// GEMMs. Attention K/V chunks are staged global->LDS with the Tensor Data
// Mover (TENSOR_LOAD_TO_LDS + s_wait_tensorcnt), removing the strided
// per-lane loads from the WMMA inner loops. Inputs are never mutated: the 4
// fresh kv/pe rows live in workspace and are patched into the LDS tile.
// ---------------------------------------------------------------------------

#define B_      8
#define S_      4
#define START_  4092
#define DIM_    7168
#define H_      128
#define QLR_    1536
#define KVLR_   512
#define NOPE_   128
#define ROPE_   64
#define VH_     128
#define QKH_    192
#define MAXSEQ_ 4096
#define END_    (START_ + S_)       // 4096
#define NROW_   (B_ * S_)           // 32
#define QCOLS_  (H_ * QKH_)         // 24576
#define OCOLS_  (H_ * VH_)          // 16384
#define KTOT_   (KVLR_ + ROPE_)     // 576
#define ACH_    32                  // attention t-chunk staged in LDS
#define EPS_    1e-6f
#define SCALE_  0.07216878364870323f   // 192^-0.5

typedef float v2f __attribute__((ext_vector_type(2)));
typedef float v8f __attribute__((ext_vector_type(8)));
typedef unsigned int v4u __attribute__((ext_vector_type(4)));
typedef int v4i __attribute__((ext_vector_type(4)));
typedef int v8i __attribute__((ext_vector_type(8)));

__device__ __forceinline__ v8f vzero8() {
  v8f v;
#pragma unroll
  for (int i = 0; i < 8; ++i) v[i] = 0.f;
  return v;
}

// D(16x16,f32) = A(16x4,f32) x B(4x16,f32) + C
__device__ __forceinline__ v8f wmma4(v2f a, v2f b, v8f c) {
  return __builtin_amdgcn_wmma_f32_16x16x4_f32(
      /*neg_a=*/false, a, /*neg_b=*/false, b,
      /*c_mod=*/(short)0, c, /*reuse_a=*/false, /*reuse_b=*/false);
}

// Low 32 bits of a generic LDS pointer are the LDS byte offset (flat aperture).
__device__ __forceinline__ unsigned int lds_offset_of(const void* p) {
  return (unsigned int)(unsigned long long)p;
}

// Tensor Data Mover: DMA a 2D tile of f32 (rows x cols) from global memory
// into LDS. D# per CDNA5 ISA ch.8: group0 = count/lds_addr/global_addr/type,
// group1 = mask/data_size/dims/strides, groups 2-3 zero (2-D tensor).
__device__ __forceinline__ void tdm_load_2d_f32(unsigned int lds_off,
                                                const float* gptr,
                                                int rows, int cols,
                                                int row_stride_elts)
{
  const unsigned long long ga = (unsigned long long)gptr;
  v4u g0;
  g0[0] = 1u;                                        // count=1, user mode
  g0[1] = lds_off;                                   // lds_addr (bytes)
  g0[2] = (unsigned int)(ga & 0xffffffffu);          // global_addr[31:0]
  g0[3] = (unsigned int)((ga >> 32) & 0x01ffffffu)   // global_addr[56:32]
        | (2u << 30);                                // type = 2 ("image")
  v8i g1;
  g1[0] = (int)(2u << 16);                           // wg_mask=0, data_size=2 (4B)
  g1[1] = (int)(((unsigned)cols & 0xffffu) << 16);   // tensor_dim0[15:0]
  g1[2] = (int)((((unsigned)cols >> 16) & 0xffffu)   // tensor_dim0[31:16]
        | (((unsigned)rows & 0xffffu) << 16));       // tensor_dim1[15:0]
  g1[3] = (int)((((unsigned)rows >> 16) & 0xffffu)   // tensor_dim1[31:16]
        | (((unsigned)cols & 0xffffu) << 16));       // tile_dim0 = cols
  g1[4] = rows & 0xffff;                             // tile_dim1 = rows (tile_dim2=0)
  g1[5] = row_stride_elts;                           // tensor_dim0_stride[31:0]
  g1[6] = 0;                                         // stride hi / dim1_stride lo
  g1[7] = 0;
  const v4i z4 = {0, 0, 0, 0};
#if __clang_major__ >= 23
  const v8i z8 = {0, 0, 0, 0, 0, 0, 0, 0};
  __builtin_amdgcn_tensor_load_to_lds(g0, g1, z4, z4, z8, 0);
#else
  __builtin_amdgcn_tensor_load_to_lds(g0, g1, z4, z4, 0);
#endif
}

// ---------------------------------------------------------------- K1: x -> q_lat (rmsnorm), kv_new (rmsnorm), pe_new (rope)
__global__ __launch_bounds__(512) void k_proj_a(
    const float* __restrict__ x,
    const float* __restrict__ wq_a, const float* __restrict__ wkv_a,
    const float* __restrict__ q_norm_w, const float* __restrict__ kv_norm_w,
    const float* __restrict__ cosb, const float* __restrict__ sinb,
    float* __restrict__ q_lat, float* __restrict__ kv_new, float* __restrict__ pe_new)
{
  __shared__ float xs[DIM_];
  __shared__ float pe_raw[ROPE_];
  __shared__ float ssq[2];
  const int row = blockIdx.x;           // row = b*S + s
  const int s   = row & (S_ - 1);
  const int tid = threadIdx.x;
  const float* xr = x + (size_t)row * DIM_;
  for (int i = tid; i < DIM_; i += 512) xs[i] = xr[i];
  if (tid < 2) ssq[tid] = 0.f;
  __syncthreads();

  float acc[5];
#pragma unroll
  for (int i = 0; i < 5; ++i) acc[i] = 0.f;
  for (int k = 0; k < DIM_; ++k) {
    const float xv = xs[k];
    const float* wq_row  = wq_a  + (size_t)k * QLR_;
    const float* wkv_row = wkv_a + (size_t)k * (KVLR_ + ROPE_);
#pragma unroll
    for (int i = 0; i < 5; ++i) {
      const int j = tid + i * 512;
      if (j < QLR_ + KVLR_ + ROPE_) {
        const float w = (j < QLR_) ? wq_row[j] : wkv_row[j - QLR_];
        acc[i] = fmaf(xv, w, acc[i]);
      }
    }
  }

  float sq_q = 0.f, sq_kv = 0.f;
#pragma unroll
  for (int i = 0; i < 5; ++i) {
    const int j = tid + i * 512;
    if (j < QLR_)                       sq_q  += acc[i] * acc[i];
    else if (j < QLR_ + KVLR_)          sq_kv += acc[i] * acc[i];
    else if (j < QLR_ + KVLR_ + ROPE_)  pe_raw[j - QLR_ - KVLR_] = acc[i];
  }
  atomicAdd(&ssq[0], sq_q);
  atomicAdd(&ssq[1], sq_kv);
  __syncthreads();
  const float rq  = rsqrtf(ssq[0] / (float)QLR_  + EPS_);
  const float rkv = rsqrtf(ssq[1] / (float)KVLR_ + EPS_);
#pragma unroll
  for (int i = 0; i < 5; ++i) {
    const int j = tid + i * 512;
    if (j < QLR_)
      q_lat[(size_t)row * QLR_ + j] = acc[i] * rq * q_norm_w[j];
    else if (j < QLR_ + KVLR_)
      kv_new[(size_t)row * KVLR_ + (j - QLR_)] = acc[i] * rkv * kv_norm_w[j - QLR_];
  }
  if (tid < ROPE_ / 2) {
    const float c  = cosb[s * (ROPE_ / 2) + tid];
    const float sn = sinb[s * (ROPE_ / 2) + tid];
    const float a0 = pe_raw[2 * tid], a1 = pe_raw[2 * tid + 1];
    pe_new[(size_t)row * ROPE_ + 2 * tid]     = a0 * c - a1 * sn;
    pe_new[(size_t)row * ROPE_ + 2 * tid + 1] = a0 * sn + a1 * c;
  }
}

// ---------------------------------------------------------------- K2: q = q_lat @ wq_b   (32 x 1536 x 24576, WMMA)
__global__ __launch_bounds__(128) void k_qproj(
    const float* __restrict__ q_lat, const float* __restrict__ wq_b,
    float* __restrict__ q)
{
  const int tid = threadIdx.x;
  const int w = tid >> 5, lane = tid & 31, lo = lane & 15, hi = lane >> 4;
  const int nt = blockIdx.x * 2 + (w >> 1);
  const int m0 = (w & 1) * 16;
  const int n0 = nt * 16;
  v8f c = vzero8();
  const float* arow = q_lat + (size_t)(m0 + lo) * QLR_ + 2 * hi;
  const float* bcol = wq_b + (size_t)hi * QCOLS_ + n0 + lo;
  for (int k = 0; k < QLR_; k += 4) {
    v2f a = *(const v2f*)(arow + k);
    v2f b; b.x = bcol[(size_t)k * QCOLS_];
           b.y = bcol[(size_t)(k + 2) * QCOLS_];
    c = wmma4(a, b, c);
  }
#pragma unroll
  for (int r = 0; r < 8; ++r)
    q[(size_t)(m0 + r + 8 * hi) * QCOLS_ + n0 + lo] = c[r];
}

// ---------------------------------------------------------------- K3: rope q_pe in place
__global__ __launch_bounds__(256) void k_rope_q(
    float* __restrict__ q, const float* __restrict__ cosb, const float* __restrict__ sinb)
{
  const int idx = blockIdx.x * 256 + threadIdx.x;   // NROW_*H_*32 threads
  const int p   = idx & 31;
  const int h   = (idx >> 5) & (H_ - 1);
  const int row = idx >> 12;
  const int s   = row & (S_ - 1);
  float* base = q + (size_t)row * QCOLS_ + h * QKH_ + NOPE_ + 2 * p;
  const float a0 = base[0], a1 = base[1];
  const float c = cosb[s * 32 + p], sn = sinb[s * 32 + p];
  base[0] = a0 * c - a1 * sn;
  base[1] = a0 * sn + a1 * c;
}

// ---------------------------------------------------------------- K4: q_eff[row,h,:] = q_nope[row,h,:] @ wkv_b_w[h,:128,:]
__global__ __launch_bounds__(128) void k_absorb(
    const float* __restrict__ q, const float* __restrict__ wkv_b_w,
    float* __restrict__ q_eff)
{
  const int tid = threadIdx.x;
  const int w = tid >> 5, lane = tid & 31, lo = lane & 15, hi = lane >> 4;
  const int h  = blockIdx.x >> 4;
  const int nt = (blockIdx.x & 15) * 2 + (w >> 1);
  const int m0 = (w & 1) * 16;
  const int n0 = nt * 16;
  v8f c = vzero8();
  const float* arow = q + (size_t)(m0 + lo) * QCOLS_ + h * QKH_ + 2 * hi;
  const float* bcol = wkv_b_w + ((size_t)h * (NOPE_ + VH_) + hi) * KVLR_ + n0 + lo;
  for (int k = 0; k < NOPE_; k += 4) {
    v2f a = *(const v2f*)(arow + k);
    v2f b; b.x = bcol[(size_t)k * KVLR_];
           b.y = bcol[(size_t)(k + 2) * KVLR_];
    c = wmma4(a, b, c);
  }
#pragma unroll
  for (int r = 0; r < 8; ++r)
    q_eff[((size_t)(m0 + r + 8 * hi) * H_ + h) * KVLR_ + n0 + lo] = c[r];
}

// ---------------------------------------------------------------- K5: flash attention in the 512-d latent space
// Block = (batch b, group of 4 heads). M-tile = 4 heads x 4 query positions.
// Each 32-row t-chunk of kv/pe is DMA'd into LDS by the Tensor Data Mover.
__global__ __launch_bounds__(256) void k_attn(
    const float* __restrict__ q, const float* __restrict__ q_eff,
    const float* __restrict__ kv_cache, const float* __restrict__ pe_cache,
    const float* __restrict__ kv_new, const float* __restrict__ pe_new,
    float* __restrict__ o_lat)
{
  __shared__ float qs[16][KTOT_];        // 36.9 KB: scaled queries
  __shared__ float kbuf[ACH_][KVLR_];    // 64 KB:   kv chunk (TDM target)
  __shared__ float pbuf[ACH_][ROPE_];    // 8 KB:    pe chunk (TDM target)
  __shared__ float ps[16][ACH_];         // 2 KB:    probabilities
  __shared__ float m_run[16], l_run[16], f_cor[16];

  const int b  = blockIdx.x >> 5;        // batch
  const int hg = blockIdx.x & 31;        // head-group of 4
  const int tid = threadIdx.x;
  const int w = tid >> 5, lane = tid & 31, lo = lane & 15, hi = lane >> 4;
  (void)lane;

  for (int i = tid; i < 16 * KTOT_; i += 256) {
    const int m = i / KTOT_, c = i % KTOT_;
    const int hh = m >> 2, s = m & 3;
    const int h = hg * 4 + hh, row = b * S_ + s;
    float v;
    if (c < KVLR_) v = q_eff[((size_t)row * H_ + h) * KVLR_ + c];
    else           v = q[(size_t)row * QCOLS_ + h * QKH_ + NOPE_ + (c - KVLR_)];
    qs[m][c] = v * SCALE_;
  }
  if (tid < 16) { m_run[tid] = -__builtin_huge_valf(); l_run[tid] = 0.f; }
  __syncthreads();

  v8f oacc[4];
#pragma unroll
  for (int i = 0; i < 4; ++i) oacc[i] = vzero8();
  const int ncol0 = w * 64;              // this wave owns o columns [ncol0, ncol0+64)

  for (int t0 = 0; t0 < END_; t0 += ACH_) {
    // ---- stage chunk: wave 0 issues two TDM tile loads, waits TENSORcnt
    if (w == 0) {
      tdm_load_2d_f32(lds_offset_of(&kbuf[0][0]),
                      kv_cache + ((size_t)b * MAXSEQ_ + t0) * KVLR_,
                      ACH_, KVLR_, KVLR_);
      tdm_load_2d_f32(lds_offset_of(&pbuf[0][0]),
                      pe_cache + ((size_t)b * MAXSEQ_ + t0) * ROPE_,
                      ACH_, ROPE_, ROPE_);
      __builtin_amdgcn_s_wait_tensorcnt(0);
    }
    __syncthreads();

    // ---- patch rows belonging to the current step (cache must not be mutated)
    if (t0 + ACH_ > START_) {
      for (int i = tid; i < S_ * KVLR_; i += 256) {
        const int sr = i / KVLR_, c = i % KVLR_;
        kbuf[START_ - t0 + sr][c] = kv_new[((size_t)b * S_ + sr) * KVLR_ + c];
      }
      for (int i = tid; i < S_ * ROPE_; i += 256) {
        const int sr = i / ROPE_, c = i % ROPE_;
        pbuf[START_ - t0 + sr][c] = pe_new[((size_t)b * S_ + sr) * ROPE_ + c];
      }
      __syncthreads();
    }

    // ---- phase 1: waves 0..1 each compute one 16x16 score tile from LDS
    if (w < ACH_ / 16) {
      const int tr = w * 16 + lo;        // t row inside chunk
      v8f sc = vzero8();
      for (int k = 0; k < KTOT_; k += 4) {
        v2f a; a.x = qs[lo][k + 2 * hi]; a.y = qs[lo][k + 2 * hi + 1];
        const int c0 = k + hi, c1 = k + 2 + hi;
        v2f bb;
        bb.x = (c0 < KVLR_) ? kbuf[tr][c0] : pbuf[tr][c0 - KVLR_];
        bb.y = (c1 < KVLR_) ? kbuf[tr][c1] : pbuf[tr][c1 - KVLR_];
        sc = wmma4(a, bb, sc);
      }
#pragma unroll
      for (int r = 0; r < 8; ++r) ps[r + 8 * hi][w * 16 + lo] = sc[r];
    }
    __syncthreads();

    // ---- phase 2: online softmax update (one thread per score row)
    if (tid < 16) {
      const int m = tid;
      const int t_allow = START_ + (m & 3);   // causal bound
      float mx = m_run[m];
      for (int j = 0; j < ACH_; ++j)
        if (t0 + j <= t_allow) mx = fmaxf(mx, ps[m][j]);
      const float f = __expf(m_run[m] - mx);
      float sum = 0.f;
      for (int j = 0; j < ACH_; ++j) {
        const float p = (t0 + j <= t_allow) ? __expf(ps[m][j] - mx) : 0.f;
        ps[m][j] = p;
        sum += p;
      }
      l_run[m] = l_run[m] * f + sum;
      m_run[m] = mx;
      f_cor[m] = f;
    }
    __syncthreads();

    // ---- phase 3: O = O*f + P x K over this wave's 64 latent columns (LDS K)
#pragma unroll
    for (int nt = 0; nt < 4; ++nt) {
      const int n0 = ncol0 + nt * 16;
      v8f acc = oacc[nt];
#pragma unroll
      for (int r = 0; r < 8; ++r) acc[r] *= f_cor[r + 8 * hi];
      for (int kk = 0; kk < ACH_; kk += 4) {
        v2f a; a.x = ps[lo][kk + 2 * hi]; a.y = ps[lo][kk + 2 * hi + 1];
        v2f bb;
        bb.x = kbuf[kk + hi][n0 + lo];
        bb.y = kbuf[kk + 2 + hi][n0 + lo];
        acc = wmma4(a, bb, acc);
      }
      oacc[nt] = acc;
    }
    __syncthreads();                     // kbuf reuse next chunk
  }

  // ---- epilogue: normalize by softmax denominator and scatter
#pragma unroll
  for (int nt = 0; nt < 4; ++nt) {
    const int n0 = ncol0 + nt * 16;
#pragma unroll
    for (int r = 0; r < 8; ++r) {
      const int m = r + 8 * hi;
      const int hh = m >> 2, s = m & 3;
      const int h = hg * 4 + hh, row = b * S_ + s;
      o_lat[((size_t)row * H_ + h) * KVLR_ + n0 + lo] = oacc[nt][r] / l_run[m];
    }
  }
}

// ---------------------------------------------------------------- K6: o_head[row,h,:] = o_lat[row,h,:] @ wkv_b_w[h,128:,:]^T
__global__ __launch_bounds__(128) void k_oabsorb(
    const float* __restrict__ o_lat, const float* __restrict__ wkv_b_w,
    float* __restrict__ o_head)
{
  const int tid = threadIdx.x;
  const int w = tid >> 5, lane = tid & 31, lo = lane & 15, hi = lane >> 4;
  const int h  = blockIdx.x >> 2;
  const int nt = (blockIdx.x & 3) * 2 + (w >> 1);
  const int m0 = (w & 1) * 16;
  const int n0 = nt * 16;
  v8f c = vzero8();
  const float* arow = o_lat + ((size_t)(m0 + lo) * H_ + h) * KVLR_ + 2 * hi;
  // B[k][n] = W[h][NOPE_+n0+n][k] (contraction over trailing index)
  const float* brow = wkv_b_w + ((size_t)h * (NOPE_ + VH_) + NOPE_ + n0 + lo) * KVLR_ + hi;
  for (int k = 0; k < KVLR_; k += 4) {
    v2f a = *(const v2f*)(arow + k);
    v2f b; b.x = brow[k];
           b.y = brow[k + 2];
    c = wmma4(a, b, c);
  }
#pragma unroll
  for (int r = 0; r < 8; ++r)
    o_head[(size_t)(m0 + r + 8 * hi) * OCOLS_ + h * VH_ + n0 + lo] = c[r];
}

// ---------------------------------------------------------------- K7: out = o_head @ wo   (32 x 16384 x 7168, WMMA)
__global__ __launch_bounds__(128) void k_out(
    const float* __restrict__ o_head, const float* __restrict__ wo,
    float* __restrict__ out)
{
  const int tid = threadIdx.x;
  const int w = tid >> 5, lane = tid & 31, lo = lane & 15, hi = lane >> 4;
  const int nt = blockIdx.x * 2 + (w >> 1);
  const int m0 = (w & 1) * 16;
  const int n0 = nt * 16;
  v8f c = vzero8();
  const float* arow = o_head + (size_t)(m0 + lo) * OCOLS_ + 2 * hi;
  const float* bcol = wo + (size_t)hi * DIM_ + n0 + lo;
  for (int k = 0; k < OCOLS_; k += 4) {
    v2f a = *(const v2f*)(arow + k);
    v2f b; b.x = bcol[(size_t)k * DIM_];
           b.y = bcol[(size_t)(k + 2) * DIM_];
    c = wmma4(a, b, c);
  }
#pragma unroll
  for (int r = 0; r < 8; ++r)
    out[(size_t)(m0 + r + 8 * hi) * DIM_ + n0 + lo] = c[r];
}

// ---------------------------------------------------------------- launch
extern "C" void kernel_launch(void* const* d_in, const int* in_sizes, int n_in,
                              void* d_out, int out_size, void* d_ws, size_t ws_size,
                              hipStream_t stream)
{
  (void)in_sizes; (void)n_in; (void)out_size; (void)ws_size;
  const float* x         = (const float*)d_in[0];
  // d_in[1] = start_pos (compile-time constant START_)
  const float* cosb      = (const float*)d_in[2];
  const float* sinb      = (const float*)d_in[3];
  const float* kv_cache  = (const float*)d_in[4];
  const float* pe_cache  = (const float*)d_in[5];
  const float* wq_a      = (const float*)d_in[6];
  const float* q_norm_w  = (const float*)d_in[7];
  const float* wq_b      = (const float*)d_in[8];
  const float* wkv_a     = (const float*)d_in[9];
  const float* kv_norm_w = (const float*)d_in[10];
  const float* wkv_b_w   = (const float*)d_in[11];
  const float* wo        = (const float*)d_in[12];
  float* out = (float*)d_out;

  float* ws     = (float*)d_ws;
  float* q_lat  = ws;                                   // 32*1536
  float* q      = q_lat  + (size_t)NROW_ * QLR_;        // 32*24576
  float* q_eff  = q      + (size_t)NROW_ * QCOLS_;      // 32*128*512
  float* o_lat  = q_eff  + (size_t)NROW_ * H_ * KVLR_;  // 32*128*512
  float* o_head = o_lat  + (size_t)NROW_ * H_ * KVLR_;  // 32*16384
  float* kv_new = o_head + (size_t)NROW_ * OCOLS_;      // 32*512
  float* pe_new = kv_new + (size_t)NROW_ * KVLR_;       // 32*64

  k_proj_a <<<NROW_, 512, 0, stream>>>(x, wq_a, wkv_a, q_norm_w, kv_norm_w,
                                       cosb, sinb, q_lat, kv_new, pe_new);
  k_qproj  <<<QCOLS_ / 32, 128, 0, stream>>>(q_lat, wq_b, q);
  k_rope_q <<<(NROW_ * H_ * 32) / 256, 256, 0, stream>>>(q, cosb, sinb);
  k_absorb <<<H_ * 16, 128, 0, stream>>>(q, wkv_b_w, q_eff);
  k_attn   <<<B_ * 32, 256, 0, stream>>>(q, q_eff, kv_cache, pe_cache,
                                         kv_new, pe_new, o_lat);
  k_oabsorb<<<H_ * 4, 128, 0, stream>>>(o_lat, wkv_b_w, o_head);
  k_out    <<<DIM_ / 32, 128, 0, stream>>>(o_head, wo, out);
}